// LocalDE_5428838662290
// MI455X (gfx1250) — compile-verified
//
#include <hip/hip_runtime.h>

typedef float v2f __attribute__((ext_vector_type(2)));
typedef float v8f __attribute__((ext_vector_type(8)));

#define B_   32
#define C_   3
#define H_   64
#define W_   64
#define HOUT 62
#define WOUT 62
#define L_   (HOUT * WOUT)      // 3844 sites per batch
#define D_   27                 // C*KH*KW
#define NPAIR 378               // D*(D+1)/2

// x layout: [B, C, H, W] row-major; patch element d = c*9 + i*3 + j
__device__ __forceinline__ float load_elem(const float* __restrict__ x,
                                           int b, int oh, int ow, int d) {
    int c = d / 9, r = d % 9;
    int i = r / 3, j = r % 3;
    return x[(((size_t)b * C_ + c) * H_ + (oh + i)) * W_ + (ow + j)];
}

__global__ void __launch_bounds__(256)
poly2_outer_wmma(const float* __restrict__ x, float* __restrict__ out) {
    const int lane = threadIdx.x & 31;
    const int nl   = lane & 15;        // N within a tile / lane within half
    const int half = lane >> 4;        // 0: lanes 0-15, 1: lanes 16-31

    // one wave per (b, l); grid sized so every wave is fully populated
    const size_t w = (size_t)blockIdx.x * 8 + (threadIdx.x >> 5);
    const int b  = (int)(w / L_);
    const int l  = (int)(w % L_);
    const int oh = l / WOUT, ow = l % WOUT;

    // Gather the 27-element patch vector into the K=0 slots of A/B.
    // A 16x4 f32: VGPR0 lanes 0-15 hold (M=lane, K=0); lanes 16-31 hold K=2 -> 0.
    // B 4x16 f32: VGPR0 lanes 0-15 hold row K=0 (N=lane); other rows -> 0.
    float vlo = 0.f, vhi = 0.f;
    if (half == 0) {
        vlo = load_elem(x, b, oh, ow, lane);                 // p[0..15]
        if (nl < D_ - 16) vhi = load_elem(x, b, oh, ow, 16 + nl); // p[16..26]
    }
    v2f A0 = {vlo, 0.f};   // rows 0..15
    v2f A1 = {vhi, 0.f};   // rows 16..26 (27..31 are zero)
    v2f B0 = A0;           // cols 0..15
    v2f B1 = A1;           // cols 16..26

    v8f cz = {};
    // D = A x B with K=1 effective -> rank-1 outer product tiles of p (x) p
    v8f D00 = __builtin_amdgcn_wmma_f32_16x16x4_f32(false, A0, false, B0,
                                                    (short)0, cz, false, false);
    v8f D01 = __builtin_amdgcn_wmma_f32_16x16x4_f32(false, A0, false, B1,
                                                    (short)0, cz, false, false);
    v8f D11 = __builtin_amdgcn_wmma_f32_16x16x4_f32(false, A1, false, B1,
                                                    (short)0, cz, false, false);

    float* base = out + ((size_t)b * L_ + l) * NPAIR;

    // D tile layout: VGPR k -> (M = k + 8*half, N = nl) within the tile.
    // Pack upper triangle: idx(i,j) = i*D - i*(i-1)/2 + (j-i)
    #pragma unroll
    for (int k = 0; k < 8; ++k) {
        {   // tile (0,0): m,n in 0..15
            int m = k + 8 * half, n = nl;
            if (n >= m) {
                int idx = m * D_ - (m * (m - 1)) / 2 + (n - m);
                base[idx] = D00[k];
            }
        }
        {   // tile (0,1): m in 0..15, n in 16..26 -> always upper triangle
            int m = k + 8 * half, n = 16 + nl;
            if (n < D_) {
                int idx = m * D_ - (m * (m - 1)) / 2 + (n - m);
                base[idx] = D01[k];
            }
        }
        {   // tile (1,1): m,n in 16..26
            int m = 16 + k + 8 * half, n = 16 + nl;
            if (m < D_ && n < D_ && n >= m) {
                int idx = m * D_ - (m * (m - 1)) / 2 + (n - m);
                base[idx] = D11[k];
            }
        }
    }
}

extern "C" void kernel_launch(void* const* d_in, const int* in_sizes, int n_in,
                              void* d_out, int out_size, void* d_ws, size_t ws_size,
                              hipStream_t stream) {
    const float* x = (const float*)d_in[0];
    float* out = (float*)d_out;
    // 123,008 (b,l) waves total, 8 waves per 256-thread block -> 15,376 blocks
    const int nblocks = (B_ * L_) / 8;
    poly2_outer_wmma<<<nblocks, 256, 0, stream>>>(x, out);
}